// RBCEquationLoss_22204980920841
// MI455X (gfx1250) — compile-verified
//
#include <hip/hip_runtime.h>

typedef __attribute__((ext_vector_type(2))) float v2f;
typedef __attribute__((ext_vector_type(8))) float v8f;
typedef __attribute__((ext_vector_type(2))) int   v2i;

#if defined(__AMDGCN__) && __has_builtin(__builtin_amdgcn_global_load_async_to_lds_b64) && __has_builtin(__builtin_amdgcn_s_wait_asynccnt)
#define HAS_ASYNC 1
#else
#define HAS_ASYNC 0
#endif

#if defined(__AMDGCN__) && __has_builtin(__builtin_amdgcn_wmma_f32_16x16x4_f32)
#define HAS_WMMA 1
#else
#define HAS_WMMA 0
#endif

namespace {
constexpr int  Bn = 2, NT = 48, Hh = 256, Ww = 512;
constexpr int  NROWS = Bn * NT * Hh;                 // 24576 blocks
constexpr long long ZS = Ww;                         // x-stride of z
constexpr long long TS = (long long)Hh * Ww;         // 131072
constexpr long long CS = (long long)NT * TS;         // 6291456
constexpr long long BS = 5 * CS;

constexpr float DTv = 10.0f / 48.0f;
constexpr float DXv = 6.283185307179586f / 512.0f;
constexpr float DZv = 2.0f / 256.0f;
constexpr float IDTc = 1.0f / DTv, I2DT = 0.5f / DTv;
constexpr float IDXc = 1.0f / DXv, I2DX = 0.5f / DXv;
constexpr float IDZc = 1.0f / DZv, I2DZ = 0.5f / DZv;
constexpr float KAP = 0.01f, NUc = 0.01f;
constexpr float INV_N = 1.0f / 12582912.0f;          // 2*48*256*512
} // namespace

// ---- 8-byte global -> LDS copy (async on CDNA5, sync fallback) -------------
__device__ __forceinline__ void copy8(const float* __restrict__ src, float* dst) {
#if HAS_ASYNC
  __builtin_amdgcn_global_load_async_to_lds_b64(
      (__attribute__((address_space(1))) v2i*)src,
      (__attribute__((address_space(3))) v2i*)dst, 0, 0);
#else
  *(float2*)dst = *(const float2*)src;
#endif
}

// ---- jnp.gradient helpers (exact boundary semantics) ------------------------
__device__ __forceinline__ float ddx(const float* r, int x) {
  if (x == 0)        return (r[1] - r[0]) * IDXc;
  if (x == Ww - 1)   return (r[Ww - 1] - r[Ww - 2]) * IDXc;
  return (r[x + 1] - r[x - 1]) * I2DX;
}
__device__ __forceinline__ float ddxx(const float* r, int x) {
  if (x == 0)        return (ddx(r, 1) - ddx(r, 0)) * IDXc;
  if (x == Ww - 1)   return (ddx(r, Ww - 1) - ddx(r, Ww - 2)) * IDXc;
  return (ddx(r, x + 1) - ddx(r, x - 1)) * I2DX;
}
// z-stack rows live at lds[base + (j - z + 2)] for logical z index j
__device__ __forceinline__ float rowv(const float (*L)[Ww], int base, int j, int z, int x) {
  return L[base + (j - z + 2)][x];
}
__device__ __forceinline__ float ddz_at(const float (*L)[Ww], int base, int zz, int z, int x) {
  if (zz == 0)       return (rowv(L, base, 1, z, x) - rowv(L, base, 0, z, x)) * IDZc;
  if (zz == Hh - 1)  return (rowv(L, base, Hh - 1, z, x) - rowv(L, base, Hh - 2, z, x)) * IDZc;
  return (rowv(L, base, zz + 1, z, x) - rowv(L, base, zz - 1, z, x)) * I2DZ;
}
__device__ __forceinline__ float ddzz(const float (*L)[Ww], int base, int z, int x) {
  if (z == 0)        return (ddz_at(L, base, 1, z, x) - ddz_at(L, base, 0, z, x)) * IDZc;
  if (z == Hh - 1)   return (ddz_at(L, base, Hh - 1, z, x) - ddz_at(L, base, Hh - 2, z, x)) * IDZc;
  return (ddz_at(L, base, z + 1, z, x) - ddz_at(L, base, z - 1, z, x)) * I2DZ;
}
__device__ __forceinline__ float ddt_f(const float (*L)[Ww], int rc, int rm, int rp, int t, int x) {
  if (t == 0)        return (L[rp][x] - L[rc][x]) * IDTc;
  if (t == NT - 1)   return (L[rc][x] - L[rm][x]) * IDTc;
  return (L[rp][x] - L[rm][x]) * I2DT;
}

// ---- main fused stencil + partial reduction --------------------------------
// Rows in LDS:
//   0-4  : T  at (t, z-2..z+2)    5-9  : u same    10-14 : w same
//   15/16: T at (t-1,z)/(t+1,z)   17/18: u         19/20 : w
//   21-23: p3 at (t, z-1..z+1)    24-26: p4 same       (all z/t clamped)
__global__ __launch_bounds__(256) void rbc_rows(const float* __restrict__ pred,
                                                float* __restrict__ part) {
  __shared__ float lds[27][Ww];
  __shared__ float red[256];
  const int tid = threadIdx.x;
  const int r   = blockIdx.x;
  const int b   = r / (NT * Hh);
  const int rem = r - b * (NT * Hh);
  const int t   = rem / Hh;
  const int z   = rem - t * Hh;

  const float* base = pred + (long long)b * BS;
  const int tm = (t > 0) ? t - 1 : 0;
  const int tp = (t < NT - 1) ? t + 1 : NT - 1;
  int zc[5];
#pragma unroll
  for (int i = 0; i < 5; ++i) {
    int zz = z - 2 + i;
    zc[i] = zz < 0 ? 0 : (zz > Hh - 1 ? Hh - 1 : zz);
  }

  const int xo = tid * 2;
  auto LR = [&](int row, int c, int tt, int zz) {
    const float* src = base + (long long)c * CS + (long long)tt * TS + (long long)zz * ZS + xo;
    copy8(src, &lds[row][xo]);
  };
#pragma unroll
  for (int i = 0; i < 5; ++i) LR(i, 0, t, zc[i]);
#pragma unroll
  for (int i = 0; i < 5; ++i) LR(5 + i, 1, t, zc[i]);
#pragma unroll
  for (int i = 0; i < 5; ++i) LR(10 + i, 2, t, zc[i]);
  LR(15, 0, tm, z); LR(16, 0, tp, z);
  LR(17, 1, tm, z); LR(18, 1, tp, z);
  LR(19, 2, tm, z); LR(20, 2, tp, z);
#pragma unroll
  for (int i = 0; i < 3; ++i) LR(21 + i, 3, t, zc[1 + i]);
#pragma unroll
  for (int i = 0; i < 3; ++i) LR(24 + i, 4, t, zc[1 + i]);

#if HAS_ASYNC
  __builtin_amdgcn_s_wait_asynccnt(0);
#endif
  __syncthreads();

  float s = 0.0f;
#pragma unroll
  for (int px = 0; px < 2; ++px) {
    const int x = tid + px * 256;
    const float* Trow = lds[2];
    const float* Urow = lds[7];
    const float* Wrow = lds[12];
    const float T0 = Trow[x], u0 = Urow[x], w0 = Wrow[x];

    // l1: divergence
    const float dudx = ddx(Urow, x);
    const float dwdz = ddz_at(lds, 10, z, z, x);
    const float divg = dudx + dwdz;

    // l2: temperature equation
    const float dTdt  = ddt_f(lds, 2, 15, 16, t, x);
    const float dTdx  = ddx(Trow, x);
    const float dTdz  = ddz_at(lds, 0, z, z, x);
    const float dTdxx = ddxx(Trow, x);
    const float dTdzz = ddzz(lds, 0, z, x);
    const float r2 = (dTdt + u0 * dTdx + w0 * dTdz) - KAP * (dTdxx + dTdzz);

    // momentum pieces
    const float dudt  = ddt_f(lds, 7, 17, 18, t, x);
    const float dudz  = ddz_at(lds, 5, z, z, x);
    const float dudxx = ddxx(Urow, x);
    const float dudzz = ddzz(lds, 5, z, x);

    const float dwdt  = ddt_f(lds, 12, 19, 20, t, x);
    const float dwdx  = ddx(Wrow, x);
    const float dwdxx = ddxx(Wrow, x);
    const float dwdzz = ddzz(lds, 10, z, x);

    // pressure p = p3 + p4 (rows 21..23 / 24..26, k: 0=z-1,1=z,2=z+1 clamped)
    auto pAt = [&](int k, int xx) { return lds[21 + k][xx] + lds[24 + k][xx]; };
    float dpdx, dpdz;
    if (x == 0)            dpdx = (pAt(1, 1) - pAt(1, 0)) * IDXc;
    else if (x == Ww - 1)  dpdx = (pAt(1, Ww - 1) - pAt(1, Ww - 2)) * IDXc;
    else                   dpdx = (pAt(1, x + 1) - pAt(1, x - 1)) * I2DX;
    if (z == 0)            dpdz = (pAt(2, x) - pAt(1, x)) * IDZc;
    else if (z == Hh - 1)  dpdz = (pAt(1, x) - pAt(0, x)) * IDZc;
    else                   dpdz = (pAt(2, x) - pAt(0, x)) * I2DZ;

    const float r3 = (dudt + u0 * dudx + w0 * dudz) - (-dpdx + NUc * (dudxx + dudzz));
    const float r4 = (dwdt + u0 * dwdx + w0 * dwdz) - (-dpdz + NUc * (dwdxx + dwdzz) + T0);

    s += divg * divg + r2 * r2 + r3 * r3 + r4 * r4;
  }

  red[tid] = s;
  __syncthreads();
#pragma unroll
  for (int o = 128; o > 0; o >>= 1) {
    if (tid < o) red[tid] += red[tid + o];
    __syncthreads();
  }
  if (tid == 0) part[r] = red[0];
}

// ---- final reduction: one wave32; cross-lane sum via V_WMMA_F32_16X16X4_F32
// A = per-lane {s, 0} (16x4 f32), B = ones (4x16): D[m][n] = rowsum_m(A).
// Sum of the 8 D VGPRs per lane = rowsums 0-7 (lanes 0-15) / 8-15 (lanes 16-31);
// adding shfl_xor(16) gives the exact 32-lane total.
__global__ __launch_bounds__(32) void rbc_finish(const float* __restrict__ part,
                                                 int n, float* __restrict__ out) {
  float s = 0.0f;
  for (int i = threadIdx.x; i < n; i += 32) s += part[i];
#if HAS_WMMA
  v2f a; a.x = s; a.y = 0.0f;
  v2f ones; ones.x = 1.0f; ones.y = 1.0f;
  v8f c = {};
  c = __builtin_amdgcn_wmma_f32_16x16x4_f32(false, a, false, ones,
                                            (short)0, c, false, false);
  float p = c[0] + c[1] + c[2] + c[3] + c[4] + c[5] + c[6] + c[7];
  p += __shfl_xor(p, 16, 32);
  if (threadIdx.x == 0) out[0] = p * INV_N;
#else
  for (int off = 16; off > 0; off >>= 1) s += __shfl_down(s, off, 32);
  if (threadIdx.x == 0) out[0] = s * INV_N;
#endif
}

extern "C" void kernel_launch(void* const* d_in, const int* in_sizes, int n_in,
                              void* d_out, int out_size, void* d_ws, size_t ws_size,
                              hipStream_t stream) {
  const float* pred = (const float*)d_in[0];
  float* part = (float*)d_ws;              // NROWS floats (96 KB) of scratch
  rbc_rows<<<dim3(NROWS), dim3(256), 0, stream>>>(pred, part);
  rbc_finish<<<dim3(1), dim3(32), 0, stream>>>(part, NROWS, (float*)d_out);
}